// WassersteinFeatureExtractor_64269890617657
// MI455X (gfx1250) — compile-verified
//
#include <hip/hip_runtime.h>
#include <math.h>

// Problem constants (match reference)
#define BB 4
#define DD 512
#define NN 1024
#define KK 64
#define EPSV 1e-10f
#define LOG_CONST_F 470.4965290008f   // D/2 * ln(2*pi), D=512

typedef __attribute__((ext_vector_type(2))) float f32x2;
typedef __attribute__((ext_vector_type(8))) float f32x8;

// D = A(16x4 f32) * B(4x16 f32) + C(16x16 f32)   -> v_wmma_f32_16x16x4_f32
static __device__ __forceinline__ f32x8 wmma4(f32x2 a, f32x2 b, f32x8 c) {
    return __builtin_amdgcn_wmma_f32_16x16x4_f32(
        /*neg_a=*/false, a, /*neg_b=*/false, b,
        /*c_mod=*/(short)0, c, /*reuse_a=*/false, /*reuse_b=*/false);
}

// ---------------------------------------------------------------------------
// Kernel A: per-cluster prep, writing a PACKED weight image.
// For inner-dim pair p = d>>1, cluster m, parity q = d&1:
//   WW[(p*K + m)*4 + 0 + q] =  alpha * mu[m,d] * iv[m,d]     (W2 pair)
//   WW[(p*K + m)*4 + 2 + q] = -0.5 * alpha * iv[m,d]         (W1 pair)
// so a stage-1 lane fetches its whole A-operand (W2 pair + W1 pair) with ONE
// global_load_b128.  Also: ck[k] = bias - 0.5*alpha*Sum(mu^2*iv) - LOG_CONST.
// ---------------------------------------------------------------------------
__global__ __launch_bounds__(256) void prep_kernel(
    const float* __restrict__ mu, const float* __restrict__ logvar,
    const float* __restrict__ bias, const float* __restrict__ alpha_p,
    float* __restrict__ WW, float* __restrict__ ck)
{
    const int k   = blockIdx.x;     // cluster 0..63
    const int tid = threadIdx.x;    // 0..255
    const float alpha = alpha_p[0];
    float c3 = 0.0f;
    for (int d = tid; d < DD; d += 256) {
        float lv = logvar[k * DD + d];
        float v  = __expf(lv) + EPSV;
        float iv = 1.0f / fmaxf(v, EPSV);
        float m  = mu[k * DD + d];
        const size_t base = ((size_t)(d >> 1) * KK + k) * 4 + (d & 1);
        WW[base]     = alpha * m * iv;          // W2 (x term)
        WW[base + 2] = -0.5f * alpha * iv;      // W1 (x^2 term)
        c3 += m * m * iv;
    }
    __shared__ float red[256];
    red[tid] = c3;
    __syncthreads();
    for (int s = 128; s > 0; s >>= 1) {
        if (tid < s) red[tid] += red[tid + s];
        __syncthreads();
    }
    if (tid == 0) ck[k] = bias[k] - 0.5f * alpha * red[0] - LOG_CONST_F;
}

// ---------------------------------------------------------------------------
// Kernel B: logits via WMMA + softmax over K.
// One wave = one (batch, 32-token tile): two 64x16 output tiles so every
// packed-weight A load (b128) feeds 4 WMMAs.  Inner dim = D (step 4).
//   E = W2^T @ x + W1^T @ x^2, then +ck, max-sub, clip, softmax over K=64.
// ---------------------------------------------------------------------------
__global__ __launch_bounds__(32) void stage1_kernel(
    const float* __restrict__ x, const float* __restrict__ WW,
    const float* __restrict__ ck, float* __restrict__ prob)
{
    const int wid  = blockIdx.x;        // 0..127
    const int b    = wid >> 5;          // /32
    const int n0   = (wid & 31) << 5;   // *32
    const int lane = threadIdx.x & 31;
    const int half = lane >> 4;
    const int lq   = lane & 15;

    const float* xp = x + (size_t)b * DD * NN;

    f32x8 acc[2][4] = {};   // [token-tile][m-tile]; m-tiles cover clusters 0..63

    for (int d = 0; d < DD; d += 4) {
        const int r0 = d + 2 * half;           // this lane's two inner rows
        // B tiles: x[b, r, col] for r = r0, r0+1 and two token tiles
        const float* xr0 = xp + (size_t)r0 * NN + n0 + lq;
        float xa0 = xr0[0];
        float xa1 = xr0[NN];
        float xb0 = xr0[16];
        float xb1 = xr0[NN + 16];
        f32x2 bA  = {xa0, xa1};
        f32x2 bA2 = {xa0 * xa0, xa1 * xa1};
        f32x2 bB  = {xb0, xb1};
        f32x2 bB2 = {xb0 * xb0, xb1 * xb1};
        // A tiles: one b128 per m-tile = (W2[r0],W2[r0+1],W1[r0],W1[r0+1])
        const float4* wp = reinterpret_cast<const float4*>(WW) + (size_t)(r0 >> 1) * KK;
#pragma unroll
        for (int t = 0; t < 4; ++t) {
            float4 w = wp[(t << 4) + lq];
            f32x2 a2 = {w.x, w.y};
            f32x2 a1 = {w.z, w.w};
            acc[0][t] = wmma4(a2, bA,  acc[0][t]);
            acc[0][t] = wmma4(a1, bA2, acc[0][t]);
            acc[1][t] = wmma4(a2, bB,  acc[1][t]);
            acc[1][t] = wmma4(a1, bB2, acc[1][t]);
        }
    }

    // epilogue: +ck then softmax over the 64 clusters, per token.
    // acc[u][t][j] holds cluster k = 16t + j + 8*half, token n0 + 16u + lq.
#pragma unroll
    for (int u = 0; u < 2; ++u) {
        float mx = -1e30f;
#pragma unroll
        for (int t = 0; t < 4; ++t)
#pragma unroll
            for (int j = 0; j < 8; ++j) {
                const int k = 16 * t + j + 8 * half;
                float e = acc[u][t][j] + ck[k];
                acc[u][t][j] = e;
                mx = fmaxf(mx, e);
            }
        mx = fmaxf(mx, __shfl_xor(mx, 16, 32));   // merge the two half-waves

        float s = 0.0f;
#pragma unroll
        for (int t = 0; t < 4; ++t)
#pragma unroll
            for (int j = 0; j < 8; ++j) {
                float e = acc[u][t][j] - mx;
                e = fminf(fmaxf(e, -100.0f), 100.0f);
                float p = __expf(e);
                acc[u][t][j] = p;
                s += p;
            }
        s += __shfl_xor(s, 16, 32);
        const float inv = 1.0f / s;

#pragma unroll
        for (int t = 0; t < 4; ++t)
#pragma unroll
            for (int j = 0; j < 8; ++j) {
                const int k = 16 * t + j + 8 * half;
                prob[((size_t)(b * KK + k)) * NN + n0 + 16 * u + lq] = acc[u][t][j] * inv;
            }
    }
}

// ---------------------------------------------------------------------------
// Kernel C: rdenom[b,k] = 1 / clip(sum_n prob, EPS). One wave per (b,k) row.
// ---------------------------------------------------------------------------
__global__ __launch_bounds__(32) void denom_kernel(
    const float* __restrict__ prob, float* __restrict__ rdenom)
{
    const int row  = blockIdx.x;        // b*64 + k, 0..255
    const int lane = threadIdx.x & 31;
    const float* p = prob + (size_t)row * NN;
    float s = 0.0f;
    for (int n = lane; n < NN; n += 32) s += p[n];
#pragma unroll
    for (int off = 16; off > 0; off >>= 1) s += __shfl_xor(s, off, 32);
    if (lane == 0) rdenom[row] = 1.0f / fmaxf(s, EPSV);
}

// ---------------------------------------------------------------------------
// Kernel D: moments via WMMA + feature epilogue.
// One wave = one (batch, 32-feature tile): two d-tiles share each prob A load.
//   Sx  (64 x d-tile) = prob @ x^T,  Sxx = prob @ (x^2)^T  (inner dim N)
//   mean_off = rd*Sx - mu ; var = rd*Sxx - 2*mu*rd*Sx + mu^2
//   std_off  = sqrt(clip(var)) - sqrt(exp(logvar)+EPS)
// ---------------------------------------------------------------------------
__global__ __launch_bounds__(32) void stage2_kernel(
    const float* __restrict__ x, const float* __restrict__ prob,
    const float* __restrict__ rdenom, const float* __restrict__ mu,
    const float* __restrict__ logvar, float* __restrict__ feats)
{
    const int wid  = blockIdx.x;        // 0..63
    const int b    = wid >> 4;          // /16
    const int d0   = (wid & 15) << 5;   // *32
    const int lane = threadIdx.x & 31;
    const int half = lane >> 4;
    const int lq   = lane & 15;

    const float* xp = x + (size_t)b * DD * NN;
    const float* pb = prob + (size_t)b * KK * NN;

    f32x8 sx[2][4] = {};   // [d-tile][m-tile]
    f32x8 sq[2][4] = {};

    for (int n = 0; n < NN; n += 4) {
        const int c0 = n + 2 * half;   // this lane's two inner columns
        float2 xva = *reinterpret_cast<const float2*>(xp + (size_t)(d0 + lq) * NN + c0);
        float2 xvb = *reinterpret_cast<const float2*>(xp + (size_t)(d0 + 16 + lq) * NN + c0);
        f32x2 bA  = {xva.x, xva.y};
        f32x2 bA2 = {xva.x * xva.x, xva.y * xva.y};
        f32x2 bB  = {xvb.x, xvb.y};
        f32x2 bB2 = {xvb.x * xvb.x, xvb.y * xvb.y};
#pragma unroll
        for (int t = 0; t < 4; ++t) {
            const int m0 = t << 4;
            float2 pv = *reinterpret_cast<const float2*>(pb + (size_t)(m0 + lq) * NN + c0);
            f32x2 ap = {pv.x, pv.y};
            sx[0][t] = wmma4(ap, bA,  sx[0][t]);
            sq[0][t] = wmma4(ap, bA2, sq[0][t]);
            sx[1][t] = wmma4(ap, bB,  sx[1][t]);
            sq[1][t] = wmma4(ap, bB2, sq[1][t]);
        }
    }

#pragma unroll
    for (int u = 0; u < 2; ++u) {
        const int dd = d0 + 16 * u + lq;
#pragma unroll
        for (int t = 0; t < 4; ++t)
#pragma unroll
            for (int j = 0; j < 8; ++j) {
                const int k  = 16 * t + j + 8 * half;
                const float rd    = rdenom[b * KK + k];
                const float meanx = sx[u][t][j] * rd;   // Sum_n gamma * x
                const float ex2   = sq[u][t][j] * rd;   // Sum_n gamma * x^2
                const float m     = mu[k * DD + dd];
                const float mean_off = meanx - m;
                float var_e = ex2 - 2.0f * m * meanx + m * m;
                var_e = fmaxf(var_e, EPSV);
                const float lv = logvar[k * DD + dd];
                const float std_off = __fsqrt_rn(var_e) - __fsqrt_rn(__expf(lv) + EPSV);
                const size_t base = ((size_t)(b * KK + k)) * (2 * DD);
                feats[base + dd]      = mean_off;
                feats[base + DD + dd] = std_off;
            }
    }
}

// ---------------------------------------------------------------------------
// Kernel E: per-batch L2 norm of the 65536-element feature row.
// ---------------------------------------------------------------------------
__global__ __launch_bounds__(256) void norm_kernel(
    const float* __restrict__ feats, float* __restrict__ scale)
{
    const int b   = blockIdx.x;
    const int tid = threadIdx.x;
    const float* f = feats + (size_t)b * (KK * 2 * DD);
    float s = 0.0f;
    for (int i = tid; i < KK * 2 * DD; i += 256) {
        float v = f[i];
        s += v * v;
    }
    __shared__ float red[256];
    red[tid] = s;
    __syncthreads();
    for (int st = 128; st > 0; st >>= 1) {
        if (tid < st) red[tid] += red[tid + st];
        __syncthreads();
    }
    if (tid == 0) scale[b] = 1.0f / fmaxf(__fsqrt_rn(red[0]), 1e-12f);
}

// ---------------------------------------------------------------------------
// Kernel F: scale into output.
// ---------------------------------------------------------------------------
__global__ __launch_bounds__(256) void scale_kernel(
    const float* __restrict__ feats, const float* __restrict__ scale,
    float* __restrict__ out)
{
    const int i = blockIdx.x * 256 + threadIdx.x;   // 0..262143
    out[i] = feats[i] * scale[i >> 16];             // 65536 feats per batch
}

// ---------------------------------------------------------------------------
extern "C" void kernel_launch(void* const* d_in, const int* in_sizes, int n_in,
                              void* d_out, int out_size, void* d_ws, size_t ws_size,
                              hipStream_t stream) {
    const float* x      = (const float*)d_in[0];   // (B, D, N)
    const float* mu     = (const float*)d_in[1];   // (K, D)
    const float* logvar = (const float*)d_in[2];   // (K, D)
    const float* bias   = (const float*)d_in[3];   // (K,)
    const float* alpha  = (const float*)d_in[4];   // scalar
    float* out = (float*)d_out;                    // (B, 2*K*D)

    float* ws = (float*)d_ws;
    float* WW     = ws;                      // packed weights: D*K*2 = 65536
    float* ck     = WW + DD * KK * 2;        // 64
    float* prob   = ck + KK;                 // B*K*N = 262144
    float* rdenom = prob + BB * KK * NN;     // 256
    float* feats  = rdenom + BB * KK;        // B*K*2D = 262144
    float* scalev = feats + BB * KK * 2 * DD; // 4
    // total ~2.26 MB of workspace

    prep_kernel  <<<KK, 256, 0, stream>>>(mu, logvar, bias, alpha, WW, ck);
    stage1_kernel<<<BB * (NN / 32), 32, 0, stream>>>(x, WW, ck, prob);
    denom_kernel <<<BB * KK, 32, 0, stream>>>(prob, rdenom);
    stage2_kernel<<<BB * (DD / 32), 32, 0, stream>>>(x, prob, rdenom, mu, logvar, feats);
    norm_kernel  <<<BB, 256, 0, stream>>>(feats, scalev);
    scale_kernel <<<(BB * KK * 2 * DD) / 256, 256, 0, stream>>>(feats, scalev, out);
}